// Memory_16295105921446
// MI455X (gfx1250) — compile-verified
//
#include <hip/hip_runtime.h>
#include <hip/hip_bf16.h>
#include <math.h>
#include <stdint.h>

// Sizes from the reference
#define BS 32
#define N  2048
#define W  128
#define R  8
#define EPSF 1e-8f
#define KC 64            // K-chunk staged in LDS per double-buffer phase

typedef __attribute__((ext_vector_type(2))) float v2f;
typedef __attribute__((ext_vector_type(8))) float v8f;

// ---------------- workspace layout (floats) ----------------
static constexpr size_t OFF_MEMNORM = 0;
static constexpr size_t OFF_ALLOC   = OFF_MEMNORM + N;
static constexpr size_t OFF_WW      = OFF_ALLOC   + (size_t)BS * N;
static constexpr size_t OFF_MEMNEW  = OFF_WW      + (size_t)BS * N;
static constexpr size_t OFF_MNORM2  = OFF_MEMNEW  + (size_t)N * W;
static constexpr size_t OFF_E       = OFF_MNORM2  + N;
static constexpr size_t OFF_M       = OFF_E       + (size_t)BS * N * 16;
static constexpr size_t OFF_U       = OFF_M       + (size_t)BS * N * 16;
static constexpr size_t OFF_CR      = OFF_U       + (size_t)BS * N * 16;
static constexpr size_t OFF_RW      = OFF_CR      + (size_t)BS * N * R;
static constexpr size_t OFF_C       = OFF_RW      + (size_t)BS * N * R;
static constexpr size_t OFF_D       = OFF_C       + (size_t)BS * R;

// ---- CDNA5 async memory->LDS helpers (inline asm: portable across toolchains) ----
__device__ __forceinline__ void asyncLoadB128(const float* g, const float* ldsPtr) {
    // low 32 bits of a generic pointer to __shared__ == LDS byte offset
    uint32_t lo = (uint32_t)(uintptr_t)ldsPtr;
    unsigned long long ga = (unsigned long long)(uintptr_t)g;
    asm volatile("global_load_async_to_lds_b128 %0, %1, off"
                 :: "v"(lo), "v"(ga) : "memory");
}
__device__ __forceinline__ void asyncWait0() {
    asm volatile("s_wait_asynccnt 0x0" ::: "memory");
}

// ---------------- row L2 norms of an (N x W) matrix ----------------
__global__ void kRowNorm(const float* __restrict__ src, float* __restrict__ dst) {
    __shared__ float red[W];
    int n = blockIdx.x, t = threadIdx.x;
    float v = src[(size_t)n * W + t];
    red[t] = v * v;
    __syncthreads();
    for (int s = W / 2; s > 0; s >>= 1) {
        if (t < s) red[t] += red[t + s];
        __syncthreads();
    }
    if (t == 0) dst[n] = sqrtf(red[0]);
}

// ---------------- allocation weighting (bitonic sort + cumprod) ----------------
__global__ void kSortAlloc(const float* __restrict__ usage, float* __restrict__ alloc) {
    __shared__ float vals[N];
    __shared__ int   idx[N];
    __shared__ float cp[N];
    int b = blockIdx.x, t = threadIdx.x;          // 1024 threads
    for (int i = t; i < N; i += 1024) { vals[i] = usage[(size_t)b * N + i]; idx[i] = i; }
    __syncthreads();
    for (int k = 2; k <= N; k <<= 1) {
        for (int j = k >> 1; j > 0; j >>= 1) {
            int i = ((t & ~(j - 1)) << 1) | (t & (j - 1));
            int l = i | j;
            bool up = ((i & k) == 0);
            float vi = vals[i], vl = vals[l];
            if (up ? (vi > vl) : (vi < vl)) {
                vals[i] = vl; vals[l] = vi;
                int ti = idx[i]; idx[i] = idx[l]; idx[l] = ti;
            }
            __syncthreads();
        }
    }
    if (t == 0) {                                  // exclusive cumprod (serial, tiny)
        float p = 1.0f;
        for (int i = 0; i < N; ++i) { cp[i] = p; p *= vals[i]; }
    }
    __syncthreads();
    for (int i = t; i < N; i += 1024) cp[i] = (1.0f - vals[i]) * cp[i];   // alloc_sorted
    __syncthreads();
    // reference does a gather: alloc[i] = alloc_sorted[indices[i]]
    for (int i = t; i < N; i += 1024) alloc[(size_t)b * N + i] = cp[idx[i]];
}

// ---------------- write content weighting -> ww ----------------
__global__ void kWriteWeight(const float* __restrict__ mem, const float* __restrict__ memnorm,
                             const float* __restrict__ wkey, const float* __restrict__ wstr,
                             const float* __restrict__ alloc, const float* __restrict__ ag,
                             const float* __restrict__ wg, float* __restrict__ ww) {
    __shared__ float skey[W];
    __shared__ float red[256];
    __shared__ float sim[N];
    int b = blockIdx.x, t = threadIdx.x;           // 256 threads
    if (t < W) skey[t] = wkey[(size_t)b * W + t];
    __syncthreads();
    red[t] = (t < W) ? skey[t] * skey[t] : 0.0f;
    __syncthreads();
    for (int s = 128; s > 0; s >>= 1) { if (t < s) red[t] += red[t + s]; __syncthreads(); }
    float wnorm = sqrtf(red[0]);
    float wsv = wstr[b];
    __syncthreads();

    float amax = -1e30f;
    for (int n = t; n < N; n += 256) {
        const float* row = mem + (size_t)n * W;
        float ip = 0.0f;
        for (int w = 0; w < W; ++w) ip += row[w] * skey[w];
        float s = ip / fmaxf(memnorm[n] * wnorm, EPSF) * wsv;
        sim[n] = s;
        amax = fmaxf(amax, s);
    }
    red[t] = amax; __syncthreads();
    for (int s = 128; s > 0; s >>= 1) { if (t < s) red[t] = fmaxf(red[t], red[t + s]); __syncthreads(); }
    float mx = red[0];
    __syncthreads();
    float lsum = 0.0f;
    for (int n = t; n < N; n += 256) {
        float e = __expf(sim[n] - mx);
        sim[n] = e;
        lsum += e;
    }
    red[t] = lsum; __syncthreads();
    for (int s = 128; s > 0; s >>= 1) { if (t < s) red[t] += red[t + s]; __syncthreads(); }
    float inv = 1.0f / red[0];
    float a = ag[b], g = wg[b];
    for (int n = t; n < N; n += 256)
        ww[(size_t)b * N + n] = g * (a * alloc[(size_t)b * N + n] + (1.0f - a) * sim[n] * inv);
}

// ---------------- memory_new = mem*(1 - mean_b ww*e) + mean_b ww*v ----------------
__global__ void kMemUpdate(const float* __restrict__ mem, const float* __restrict__ ww,
                           const float* __restrict__ er, const float* __restrict__ wv,
                           float* __restrict__ memnew) {
    int gid = blockIdx.x * blockDim.x + threadIdx.x;   // N*W
    int n = gid >> 7, w = gid & (W - 1);
    float e = 0.0f, a = 0.0f;
    for (int b = 0; b < BS; ++b) {
        float wwv = ww[(size_t)b * N + n];
        e = fmaf(wwv, er[(size_t)b * W + w], e);
        a = fmaf(wwv, wv[(size_t)b * W + w], a);
    }
    const float invB = 1.0f / (float)BS;
    memnew[gid] = mem[gid] * (1.0f - e * invB) + a * invB;
}

// ---------------- build B-panel E = [lrw | ww*lrw], plus c_r = prec.lrw, d_r = ww.lrw ----
__global__ void kBuildExt(const float* __restrict__ lrw, const float* __restrict__ ww,
                          const float* __restrict__ prec, float* __restrict__ E,
                          float* __restrict__ cvals, float* __restrict__ dvals) {
    __shared__ float red[256];
    int b = blockIdx.x, t = threadIdx.x;               // 256 threads
    float lc[R], ld[R];
    for (int r = 0; r < R; ++r) { lc[r] = 0.0f; ld[r] = 0.0f; }
    for (int j = t; j < N; j += 256) {
        float wwv = ww[(size_t)b * N + j];
        float pv  = prec[(size_t)b * N + j];
        const float* lr = lrw + ((size_t)b * N + j) * R;
        float* Ej = E + ((size_t)b * N + j) * 16;
        for (int r = 0; r < R; ++r) {
            float l = lr[r];
            Ej[r]     = l;
            Ej[8 + r] = wwv * l;
            lc[r] = fmaf(pv, l, lc[r]);
            ld[r] = fmaf(wwv, l, ld[r]);
        }
    }
    for (int r = 0; r < R; ++r) {
        red[t] = lc[r]; __syncthreads();
        for (int s = 128; s > 0; s >>= 1) { if (t < s) red[t] += red[t + s]; __syncthreads(); }
        if (t == 0) cvals[b * R + r] = red[0];
        __syncthreads();
        red[t] = ld[r]; __syncthreads();
        for (int s = 128; s > 0; s >>= 1) { if (t < s) red[t] += red[t + s]; __syncthreads(); }
        if (t == 0) dvals[b * R + r] = red[0];
        __syncthreads();
    }
}

// ====================================================================
// [M1|M2] = L @ E.
// 256 threads = 8 waves; each wave owns one 16x16 output tile (16 rows of L).
// Per K-chunk (64): async-DMA a 128x64 L tile + 64x16 E panel into LDS
// (coalesced 512B/instr), double buffered; WMMA fragments come from LDS
// with a 68-float row pitch so every ds_load_b64 hits 32 distinct banks.
// ====================================================================
__global__ __launch_bounds__(256) void kGemmRow(const float* __restrict__ L,
                                                const float* __restrict__ E,
                                                float* __restrict__ M) {
    __shared__ float As[2][128 * 68];   // 2 x 34816 B
    __shared__ float Es[2][KC * 16];    // 2 x  4096 B
    int b = blockIdx.y;
    int t = threadIdx.x;
    int wave = t >> 5, lane = t & 31;
    int m = lane & 15;                  // A row / B col / D col for this lane
    int koff = (lane >> 4) * 2;         // K pair: lanes 0-15 -> {0,1}, 16-31 -> {2,3}
    int hi = lane >> 4;
    const float* Lb = L + (size_t)b * N * N;
    const float* Eb = E + (size_t)b * N * 16;
    int rowbase = blockIdx.x * 128;

    // prologue: stage chunk 0 into buffer 0
    {
        for (int i = 0; i < 8; ++i) {
            int q = i * 256 + t;
            int row = q >> 4, cc = q & 15;      // 16 x b128 per 64-float row
            asyncLoadB128(Lb + (size_t)(rowbase + row) * N + cc * 4,
                          &As[0][row * 68 + cc * 4]);
        }
        asyncLoadB128(Eb + t * 4, &Es[0][t * 4]);
    }

    v8f acc = {};
    int cur = 0;
    for (int kc = 0; kc < N; kc += KC) {
        asyncWait0();                 // this wave's DMA for buffer `cur` done
        __syncthreads();              // everyone's DMA visible
        int nxt = kc + KC;
        if (nxt < N) {                // overlap next chunk's DMA with compute
            int nb = cur ^ 1;
            for (int i = 0; i < 8; ++i) {
                int q = i * 256 + t;
                int row = q >> 4, cc = q & 15;
                asyncLoadB128(Lb + (size_t)(rowbase + row) * N + nxt + cc * 4,
                              &As[nb][row * 68 + cc * 4]);
            }
            asyncLoadB128(Eb + (size_t)nxt * 16 + t * 4, &Es[nb][t * 4]);
        }
        const float* Aw = &As[cur][(wave * 16 + m) * 68];
        const float* Ep = &Es[cur][0];
#pragma unroll
        for (int kk = 0; kk < KC; kk += 4) {
            v2f a = *(const v2f*)(Aw + kk + koff);            // ds_load_b64, bank-clean
            v2f bm;
            bm.x = Ep[(kk + koff) * 16 + m];
            bm.y = Ep[(kk + koff + 1) * 16 + m];
            acc = __builtin_amdgcn_wmma_f32_16x16x4_f32(
                false, a, false, bm, (short)0, acc, false, false);
        }
        __syncthreads();              // done reading `cur` before it is re-filled
        cur ^= 1;
    }
    int itile = blockIdx.x * 8 + wave;
    float* Mb = M + (size_t)b * N * 16 + (size_t)itile * 256;
    for (int v = 0; v < 8; ++v)
        Mb[(size_t)(v + 8 * hi) * 16 + m] = acc[v];           // D: row v+8*hi, col m
}

// ====================================================================
// [U1|U2] = L^T @ E. Same pipeline; the staged tile is 64(K) x 128(cols),
// pitch 136 floats so lanes 0-15 (K-pair {0,1}) and 16-31 ({2,3}) land in
// disjoint bank halves.
// ====================================================================
__global__ __launch_bounds__(256) void kGemmCol(const float* __restrict__ L,
                                                const float* __restrict__ E,
                                                float* __restrict__ U) {
    __shared__ float Ks[2][KC * 136];   // 2 x 34816 B
    __shared__ float Es[2][KC * 16];
    int b = blockIdx.y;
    int t = threadIdx.x;
    int wave = t >> 5, lane = t & 31;
    int m = lane & 15;
    int koff = (lane >> 4) * 2;
    int hi = lane >> 4;
    const float* Lb = L + (size_t)b * N * N;
    const float* Eb = E + (size_t)b * N * 16;
    int cbase = blockIdx.x * 128;

    {
        for (int i = 0; i < 8; ++i) {
            int q = i * 256 + t;
            int row = q >> 5, cc = q & 31;      // 32 x b128 per 128-float row
            asyncLoadB128(Lb + (size_t)row * N + cbase + cc * 4,
                          &Ks[0][row * 136 + cc * 4]);
        }
        asyncLoadB128(Eb + t * 4, &Es[0][t * 4]);
    }

    v8f acc = {};
    int cur = 0;
    int cl = wave * 16 + m;                     // this lane's column within the tile
    for (int kc = 0; kc < N; kc += KC) {
        asyncWait0();
        __syncthreads();
        int nxt = kc + KC;
        if (nxt < N) {
            int nb = cur ^ 1;
            for (int i = 0; i < 8; ++i) {
                int q = i * 256 + t;
                int row = q >> 5, cc = q & 31;
                asyncLoadB128(Lb + (size_t)(nxt + row) * N + cbase + cc * 4,
                              &Ks[nb][row * 136 + cc * 4]);
            }
            asyncLoadB128(Eb + (size_t)nxt * 16 + t * 4, &Es[nb][t * 4]);
        }
        const float* Kp = &Ks[cur][0];
        const float* Ep = &Es[cur][0];
#pragma unroll
        for (int kk = 0; kk < KC; kk += 4) {
            v2f a;
            a.x = Kp[(kk + koff) * 136 + cl];   // A[m][k] = L[k][c]
            a.y = Kp[(kk + koff + 1) * 136 + cl];
            v2f bm;
            bm.x = Ep[(kk + koff) * 16 + m];
            bm.y = Ep[(kk + koff + 1) * 16 + m];
            acc = __builtin_amdgcn_wmma_f32_16x16x4_f32(
                false, a, false, bm, (short)0, acc, false, false);
        }
        __syncthreads();
        cur ^= 1;
    }
    int ctile = blockIdx.x * 8 + wave;
    float* Ub = U + (size_t)b * N * 16 + (size_t)ctile * 256;
    for (int v = 0; v < 8; ++v)
        Ub[(size_t)(v + 8 * hi) * 16 + m] = acc[v];
}

// ---------------- read content softmax -> cr ----------------
__global__ void kReadContent(const float* __restrict__ memnew, const float* __restrict__ mnorm,
                             const float* __restrict__ rkeys, const float* __restrict__ rstr,
                             float* __restrict__ crbuf) {
    __shared__ float skey[W * R];
    __shared__ float red[256];
    __shared__ float rn[R], rmax[R], rsum[R];
    int b = blockIdx.x, t = threadIdx.x;               // 256 threads
    for (int i = t; i < W * R; i += 256) skey[i] = rkeys[(size_t)b * W * R + i];
    __syncthreads();
    if (t < R) {
        float s = 0.0f;
        for (int w = 0; w < W; ++w) { float v = skey[w * R + t]; s = fmaf(v, v, s); }
        rn[t] = sqrtf(s);
    }
    __syncthreads();
    float lmax[R];
    for (int r = 0; r < R; ++r) lmax[r] = -1e30f;
    for (int n = t; n < N; n += 256) {
        const float* row = memnew + (size_t)n * W;
        float ip[R];
        for (int r = 0; r < R; ++r) ip[r] = 0.0f;
        for (int w = 0; w < W; ++w) {
            float mv = row[w];
            for (int r = 0; r < R; ++r) ip[r] = fmaf(mv, skey[w * R + r], ip[r]);
        }
        float mn = mnorm[n];
        float* cb = crbuf + ((size_t)b * N + n) * R;
        for (int r = 0; r < R; ++r) {
            float s = ip[r] / fmaxf(mn * rn[r], EPSF) * rstr[b * R + r];
            cb[r] = s;
            lmax[r] = fmaxf(lmax[r], s);
        }
    }
    for (int r = 0; r < R; ++r) {
        red[t] = lmax[r]; __syncthreads();
        for (int s = 128; s > 0; s >>= 1) { if (t < s) red[t] = fmaxf(red[t], red[t + s]); __syncthreads(); }
        if (t == 0) rmax[r] = red[0];
        __syncthreads();
    }
    float lsum[R];
    for (int r = 0; r < R; ++r) lsum[r] = 0.0f;
    for (int n = t; n < N; n += 256) {
        float* cb = crbuf + ((size_t)b * N + n) * R;
        for (int r = 0; r < R; ++r) {
            float e = __expf(cb[r] - rmax[r]);
            cb[r] = e;
            lsum[r] += e;
        }
    }
    for (int r = 0; r < R; ++r) {
        red[t] = lsum[r]; __syncthreads();
        for (int s = 128; s > 0; s >>= 1) { if (t < s) red[t] += red[t + s]; __syncthreads(); }
        if (t == 0) rsum[r] = red[0];
        __syncthreads();
    }
    for (int n = t; n < N; n += 256) {
        float* cb = crbuf + ((size_t)b * N + n) * R;
        for (int r = 0; r < R; ++r) cb[r] *= 1.0f / rsum[r];
    }
}

// ---------------- combine bw / cr / fw with read_modes ----------------
__global__ void kCombine(const float* __restrict__ Mbuf, const float* __restrict__ Ubuf,
                         const float* __restrict__ crbuf, const float* __restrict__ ww,
                         const float* __restrict__ prec, const float* __restrict__ cvals,
                         const float* __restrict__ dvals, const float* __restrict__ rmodes,
                         float* __restrict__ rw) {
    size_t gid = (size_t)blockIdx.x * blockDim.x + threadIdx.x;   // BS*N*R
    int r = (int)(gid & (R - 1));
    int n = (int)((gid >> 3) & (N - 1));
    int b = (int)(gid >> 14);
    float wwv = ww[(size_t)b * N + n];
    size_t base16 = ((size_t)b * N + n) * 16;
    float M1 = Mbuf[base16 + r],     M2 = Mbuf[base16 + 8 + r];
    float U1 = Ubuf[base16 + r],     U2 = Ubuf[base16 + 8 + r];
    float bw = M1 - M2 - wwv * M1 + wwv * cvals[b * R + r];
    float fw = U1 - wwv * U1 - U2 + prec[(size_t)b * N + n] * dvals[b * R + r];
    float cr = crbuf[gid];
    const float* rm = rmodes + ((size_t)b * R + r) * 3;
    rw[gid] = rm[0] * bw + rm[1] * cr + rm[2] * fw;
}

// ---------------- read_vectors[b][w][r] = sum_n memnew[n][w] * rw[b][n][r] ----------
__global__ void kReadVec(const float* __restrict__ memnew, const float* __restrict__ rw,
                         float* __restrict__ out) {
    __shared__ float srw[128 * R];
    int b = blockIdx.x, t = threadIdx.x;               // 128 threads, t == w
    float acc[R];
    for (int r = 0; r < R; ++r) acc[r] = 0.0f;
    for (int n0 = 0; n0 < N; n0 += 128) {
        for (int i = t; i < 128 * R; i += 128) srw[i] = rw[((size_t)b * N + n0) * R + i];
        __syncthreads();
        for (int nn = 0; nn < 128; ++nn) {
            float mv = memnew[(size_t)(n0 + nn) * W + t];        // coalesced across w
            for (int r = 0; r < R; ++r) acc[r] = fmaf(mv, srw[nn * R + r], acc[r]);
        }
        __syncthreads();
    }
    for (int r = 0; r < R; ++r) out[((size_t)b * W + t) * R + r] = acc[r];
}

// ---------------- host-side launch sequence ----------------
extern "C" void kernel_launch(void* const* d_in, const int* in_sizes, int n_in,
                              void* d_out, int out_size, void* d_ws, size_t ws_size,
                              hipStream_t stream) {
    const float* memory = (const float*)d_in[0];
    const float* usage  = (const float*)d_in[1];
    const float* prec   = (const float*)d_in[2];
    const float* L      = (const float*)d_in[3];
    const float* lrw    = (const float*)d_in[4];
    const float* rkeys  = (const float*)d_in[5];
    const float* rstr   = (const float*)d_in[6];
    const float* wkey   = (const float*)d_in[7];
    const float* wstr   = (const float*)d_in[8];
    const float* erase  = (const float*)d_in[9];
    const float* wvec   = (const float*)d_in[10];
    // d_in[11] = free_gates (dead code in the reference output path)
    const float* agate  = (const float*)d_in[12];
    const float* wgate  = (const float*)d_in[13];
    const float* rmodes = (const float*)d_in[14];
    float* out = (float*)d_out;

    float* ws = (float*)d_ws;
    float* memnorm = ws + OFF_MEMNORM;
    float* alloc   = ws + OFF_ALLOC;
    float* ww      = ws + OFF_WW;
    float* memnew  = ws + OFF_MEMNEW;
    float* mnorm2  = ws + OFF_MNORM2;
    float* E       = ws + OFF_E;
    float* Mbuf    = ws + OFF_M;
    float* Ubuf    = ws + OFF_U;
    float* crbuf   = ws + OFF_CR;
    float* rwbuf   = ws + OFF_RW;
    float* cvals   = ws + OFF_C;
    float* dvals   = ws + OFF_D;

    kRowNorm<<<N, W, 0, stream>>>(memory, memnorm);
    kSortAlloc<<<BS, 1024, 0, stream>>>(usage, alloc);
    kWriteWeight<<<BS, 256, 0, stream>>>(memory, memnorm, wkey, wstr, alloc, agate, wgate, ww);
    kMemUpdate<<<(N * W) / 256, 256, 0, stream>>>(memory, ww, erase, wvec, memnew);
    kRowNorm<<<N, W, 0, stream>>>(memnew, mnorm2);
    kBuildExt<<<BS, 256, 0, stream>>>(lrw, ww, prec, E, cvals, dvals);
    kGemmRow<<<dim3(N / 128, BS), 256, 0, stream>>>(L, E, Mbuf);
    kGemmCol<<<dim3(N / 128, BS), 256, 0, stream>>>(L, E, Ubuf);
    kReadContent<<<BS, 256, 0, stream>>>(memnew, mnorm2, rkeys, rstr, crbuf);
    kCombine<<<(BS * N * R) / 256, 256, 0, stream>>>(Mbuf, Ubuf, crbuf, ww, prec,
                                                     cvals, dvals, rmodes, rwbuf);
    kReadVec<<<BS, 128, 0, stream>>>(memnew, rwbuf, out);
}